// SA1_78675210928482
// MI455X (gfx1250) — compile-verified
//
#include <hip/hip_runtime.h>

// Problem: B=8, C=256, H=W=160.  One workgroup per n=b*c matrix (2048 blocks).
// Memory-bound (≈420MB @ 23.3TB/s ≈ 18us).  Tile staged once in LDS (stride
// padded to 164 floats: float4-aligned, conflict-free WMMA A-fetch and column
// reads).  Tile staging uses gfx1250 GLOBAL_LOAD_ASYNC_TO_LDS_B128 (ASYNCcnt)
// so HBM->LDS bypasses VGPRs; row mat-vec (logits) runs on
// V_WMMA_F32_16X16X4_F32 with the fc_w chunk broadcast into every B column.

#define Hh 160
#define Ww 160
#define HW 25600
#define SS 164          // padded LDS row stride in floats (SS%4==0, bank-clean)

typedef float v2f __attribute__((ext_vector_type(2)));
typedef float v8f __attribute__((ext_vector_type(8)));

#if defined(__gfx1250__) && __has_builtin(__builtin_amdgcn_global_load_async_to_lds_b128)
#define USE_ASYNC_LDS 1
// Builtin signature (from hipcc diagnostic): params are int4-vector pointers,
// addrspace(1) source and addrspace(3) dest, plus two immediate ints.
typedef int v4i_b128 __attribute__((vector_size(4 * sizeof(int))));
typedef __attribute__((address_space(1))) v4i_b128* async_gptr_t;
typedef __attribute__((address_space(3))) v4i_b128* async_lptr_t;
#else
#define USE_ASYNC_LDS 0
#endif

__launch_bounds__(256)
__global__ void SA1_attn_collapse_kernel(const float* __restrict__ x,
                                         const float* __restrict__ fcw_w,
                                         const float* __restrict__ fcw_b,
                                         const float* __restrict__ fch_w,
                                         const float* __restrict__ fch_b,
                                         float* __restrict__ out)
{
    __shared__ __align__(16) float tile[Hh * SS];   // 104,960 B
    __shared__ float fcw[Ww];
    __shared__ float logitS[Hh];
    __shared__ float alphaS[Hh];
    __shared__ float rowS[Ww];
    __shared__ float scal[4];                       // [0]=max1 [1]=sum1 [2]=sum(fch) [3]=s

    const int tid  = threadIdx.x;
    const int lane = tid & 31;
    const int wave = tid >> 5;
    const int n    = blockIdx.x;

    // ---- stage fc weights; wave1 reduces sum(fch_w) ----
    if (tid < Ww) fcw[tid] = fcw_w[tid];
    if (wave == 1) {
        float s = 0.f;
        for (int j = lane; j < Hh; j += 32) s += fch_w[j];
        for (int off = 16; off > 0; off >>= 1) s += __shfl_xor(s, off);
        if (lane == 0) scal[2] = s;
    }

    // ---- stage the 160x160 tile into padded LDS ----
    const float4* xg = reinterpret_cast<const float4*>(x + (size_t)n * HW);
#if USE_ASYNC_LDS
    // HBM -> LDS directly (no VGPR round-trip), tracked by ASYNCcnt.
    for (int i = tid; i < HW / 4; i += 256) {
        int f = i * 4;
        int h = f / Ww, w = f - h * Ww;             // Ww%4==0 -> b128 stays in-row
        __builtin_amdgcn_global_load_async_to_lds_b128(
            (async_gptr_t)(xg + i),
            (async_lptr_t)&tile[h * SS + w],
            /*offset=*/0, /*cpol=*/0);
    }
    // Barrier does NOT drain ASYNCcnt: each wave waits for its own asyncs,
    // then the barrier publishes the whole tile.
#if __has_builtin(__builtin_amdgcn_s_wait_asynccnt)
    __builtin_amdgcn_s_wait_asynccnt(0);
#else
    asm volatile("s_wait_asynccnt 0" ::: "memory");
#endif
#else
    for (int i = tid; i < HW / 4; i += 256) {
        float4 v = xg[i];
        int f = i * 4;
        int h = f / Ww, w = f - h * Ww;
        *reinterpret_cast<float4*>(&tile[h * SS + w]) = v;  // 16B aligned
    }
#endif
    __syncthreads();

    // ---- logits[h] = x[h,:]·fcw via V_WMMA_F32_16X16X4_F32 ----
    // 10 row-tiles of 16; wave v owns tiles v and v+8 (wave-uniform, EXEC all 1s).
    // A 16x4 f32 layout (ISA 7.12.2): lanes<16 hold K={0,1}, lanes>=16 K={2,3}.
    // B 4x16: every column = fcw chunk => every D column = logit chunk.
    for (int t = wave; t < 10; t += 8) {
        const int h0   = t * 16;
        const int L    = lane & 15;
        const int koff = (lane >= 16) ? 2 : 0;
        const float* arow = &tile[(h0 + L) * SS];
        v8f c = {0.f, 0.f, 0.f, 0.f, 0.f, 0.f, 0.f, 0.f};
        for (int k = 0; k < Ww; k += 4) {
            v2f a, b;
            a.x = arow[k + koff];
            a.y = arow[k + koff + 1];
            b.x = fcw[k + koff];
            b.y = fcw[k + koff + 1];
            c = __builtin_amdgcn_wmma_f32_16x16x4_f32(false, a, false, b,
                                                      (short)0, c, false, false);
        }
        // D: vgpr i = D[M=i, N=lane] (lanes<16) / D[M=i+8, N=lane-16] (lanes>=16)
        float bias = fcw_b[0];                       // shifts all logits equally
        if (lane == 0)
            for (int i = 0; i < 8; ++i) logitS[h0 + i] = c[i] + bias;
        if (lane == 16)
            for (int i = 0; i < 8; ++i) logitS[h0 + 8 + i] = c[i] + bias;
    }
    __syncthreads();

    // ---- softmax over h (wave0 computes max & sum, all threads form alpha) ----
    if (wave == 0) {
        float m = -3.402823466e38f;
        for (int j = lane; j < Hh; j += 32) m = fmaxf(m, logitS[j]);
        for (int off = 16; off > 0; off >>= 1) m = fmaxf(m, __shfl_xor(m, off));
        float ssum = 0.f;
        for (int j = lane; j < Hh; j += 32) ssum += __expf(logitS[j] - m);
        for (int off = 16; off > 0; off >>= 1) ssum += __shfl_xor(ssum, off);
        if (lane == 0) { scal[0] = m; scal[1] = ssum; }
    }
    __syncthreads();
    if (tid < Hh) alphaS[tid] = __expf(logitS[tid] - scal[0]) / scal[1];
    __syncthreads();

    // ---- row[w] = sum_h alpha[h] * x[h,w]  (lanes read consecutive banks) ----
    if (tid < Ww) {
        float acc = 0.f;
        const float* col = &tile[tid];
        for (int h = 0; h < Hh; ++h) acc = fmaf(alphaS[h], col[h * SS], acc);
        rowS[tid] = acc;
    }
    __syncthreads();

    // ---- beta = softmax_w(row*sum(fch) [+ fch_b, cancels]); s = beta·row ----
    if (wave == 0) {
        const float sf = scal[2];
        (void)fch_b;                                 // bias shifts logits uniformly
        float m = -3.402823466e38f;
        for (int j = lane; j < Ww; j += 32) m = fmaxf(m, rowS[j] * sf);
        for (int off = 16; off > 0; off >>= 1) m = fmaxf(m, __shfl_xor(m, off));
        float ssum = 0.f, sacc = 0.f;
        for (int j = lane; j < Ww; j += 32) {
            float e = __expf(rowS[j] * sf - m);
            ssum += e;
            sacc = fmaf(e, rowS[j], sacc);
        }
        for (int off = 16; off > 0; off >>= 1) {
            ssum += __shfl_xor(ssum, off);
            sacc += __shfl_xor(sacc, off);
        }
        if (lane == 0) scal[3] = sacc / ssum;
    }
    __syncthreads();

    // ---- broadcast s to the whole (h,w) plane: coalesced float4 stores ----
    const float sval = scal[3];
    float4 o; o.x = sval; o.y = sval; o.z = sval; o.w = sval;
    float4* og = reinterpret_cast<float4*>(out + (size_t)n * HW);
    for (int i = tid; i < HW / 4; i += 256) og[i] = o;
}

extern "C" void kernel_launch(void* const* d_in, const int* in_sizes, int n_in,
                              void* d_out, int out_size, void* d_ws, size_t ws_size,
                              hipStream_t stream) {
    const float* x     = (const float*)d_in[0];
    const float* fcw_w = (const float*)d_in[1];
    const float* fcw_b = (const float*)d_in[2];
    const float* fch_w = (const float*)d_in[3];
    const float* fch_b = (const float*)d_in[4];
    float* out = (float*)d_out;

    const int n_mats = in_sizes[0] / HW;   // B*C = 2048
    SA1_attn_collapse_kernel<<<n_mats, 256, 0, stream>>>(x, fcw_w, fcw_b,
                                                         fch_w, fch_b, out);
}